// StaticGNN_78194174591508
// MI455X (gfx1250) — compile-verified
//
#include <hip/hip_runtime.h>

#define N_NODES 50000
#define N_EDGES 600000
#define DIM 128

typedef __attribute__((ext_vector_type(2))) float v2f;
typedef __attribute__((ext_vector_type(8))) float v8f;

// ---------------------------------------------------------------------------
// degree: deg[dst[e]] += 1 over all edges
// ---------------------------------------------------------------------------
__global__ void __launch_bounds__(256)
degree_kernel(const int* __restrict__ dst, float* __restrict__ deg, int ne) {
  int e = blockIdx.x * blockDim.x + threadIdx.x;
  if (e < ne) atomicAdd(&deg[dst[e]], 1.0f);
}

// dinv = rsqrt(deg + 1)  (in place)
__global__ void __launch_bounds__(256)
dinv_kernel(float* __restrict__ deg, int n) {
  int i = blockIdx.x * blockDim.x + threadIdx.x;
  if (i < n) deg[i] = rsqrtf(deg[i] + 1.0f);
}

// ---------------------------------------------------------------------------
// Y[row0:row0+16, :] = X[row0:row0+16, :] @ W   (fp32 WMMA 16x16x4)
// block = 256 threads = 8 waves; wave w owns output columns [16w, 16w+16)
// ---------------------------------------------------------------------------
__global__ void __launch_bounds__(256)
gemm_wmma_f32(const float* __restrict__ X, const float* __restrict__ W,
              float* __restrict__ Y) {
  __shared__ float Xs[16 * 132];                 // pitch 132 avoids bank conflicts
  const int row0 = blockIdx.x * 16;

  for (int i = threadIdx.x; i < 16 * 128; i += 256) {
    int r = i >> 7, c = i & 127;
    Xs[r * 132 + c] = X[(size_t)(row0 + r) * DIM + c];
  }
  __syncthreads();

  const int wave = threadIdx.x >> 5;
  const int lane = threadIdx.x & 31;
  const int col0 = wave * 16;
  const int m    = lane & 15;
  const int kg   = lane >> 4;                    // 0 -> K{0,1}, 1 -> K{2,3}

  v8f c = {};
  const float* xrow = &Xs[m * 132];
#pragma unroll 4
  for (int k0 = 0; k0 < DIM; k0 += 4) {
    v2f a, b;
    a.x = xrow[k0 + 2 * kg + 0];
    a.y = xrow[k0 + 2 * kg + 1];
    const float* wrow = W + (size_t)(k0 + 2 * kg) * DIM + col0 + m;
    b.x = wrow[0];
    b.y = wrow[DIM];
    // D = A(16x4) * B(4x16) + C   -> v_wmma_f32_16x16x4_f32
    c = __builtin_amdgcn_wmma_f32_16x16x4_f32(false, a, false, b,
                                              (short)0, c, false, false);
  }

  // C/D layout: VGPR v -> M = v (lanes 0-15) / M = v+8 (lanes 16-31), N = lane%16
  float* yp = Y + (size_t)(row0 + kg * 8) * DIM + col0 + m;
#pragma unroll
  for (int v = 0; v < 8; ++v)
    yp[(size_t)v * DIM] = c[v];
}

// ---------------------------------------------------------------------------
// one wave per edge: agg[dst] += xw[src] * (dinv[src]*dinv[dst])
// ---------------------------------------------------------------------------
__global__ void __launch_bounds__(256)
scatter_kernel(const float* __restrict__ XW, const int* __restrict__ src,
               const int* __restrict__ dst, const float* __restrict__ dinv,
               float* __restrict__ agg) {
  const int e = blockIdx.x * 8 + (threadIdx.x >> 5);
  const int lane = threadIdx.x & 31;
  const int s = src[e];
  const int d = dst[e];
  const float nrm = dinv[s] * dinv[d];
  const float4 v = ((const float4*)(XW + (size_t)s * DIM))[lane];
  float* a = agg + (size_t)d * DIM + lane * 4;
  atomicAdd(a + 0, v.x * nrm);
  atomicAdd(a + 1, v.y * nrm);
  atomicAdd(a + 2, v.z * nrm);
  atomicAdd(a + 3, v.w * nrm);
}

// ---------------------------------------------------------------------------
// h = relu(agg + xw*dinv^2 + b + x); also re-zeroes agg for next layer
// one float4 per thread (t in [0, N*32))
// ---------------------------------------------------------------------------
__global__ void __launch_bounds__(256)
combine1_kernel(float* __restrict__ agg, const float* __restrict__ xw,
                const float* __restrict__ x, const float* __restrict__ bias,
                const float* __restrict__ dinv, float* __restrict__ h) {
  const int t = blockIdx.x * blockDim.x + threadIdx.x;
  const int node = t >> 5;
  const int j = t & 31;
  const float di = dinv[node];
  const float di2 = di * di;
  const float4 a  = ((const float4*)agg)[t];
  const float4 w  = ((const float4*)xw)[t];
  const float4 xv = ((const float4*)x)[t];
  const float4 b  = ((const float4*)bias)[j];
  float4 r;
  r.x = fmaxf(a.x + w.x * di2 + b.x + xv.x, 0.0f);
  r.y = fmaxf(a.y + w.y * di2 + b.y + xv.y, 0.0f);
  r.z = fmaxf(a.z + w.z * di2 + b.z + xv.z, 0.0f);
  r.w = fmaxf(a.w + w.w * di2 + b.w + xv.w, 0.0f);
  ((float4*)h)[t] = r;
  float4 z = {0.0f, 0.0f, 0.0f, 0.0f};
  ((float4*)agg)[t] = z;                         // reset accumulator for layer 2
}

// ---------------------------------------------------------------------------
// out[node] = relu(agg + xw2*dinv^2 + b2 + h) . Wh + bh   (one wave per node)
// ---------------------------------------------------------------------------
__global__ void __launch_bounds__(256)
combine2_head_kernel(const float* __restrict__ agg, const float* __restrict__ xw,
                     const float* __restrict__ h, const float* __restrict__ bias,
                     const float* __restrict__ dinv, const float* __restrict__ Wh,
                     const float* __restrict__ bh, float* __restrict__ out) {
  const int node = blockIdx.x * 8 + (threadIdx.x >> 5);
  const int lane = threadIdx.x & 31;
  const int t = node * 32 + lane;
  const float di = dinv[node];
  const float di2 = di * di;
  const float4 a  = ((const float4*)agg)[t];
  const float4 w  = ((const float4*)xw)[t];
  const float4 hv = ((const float4*)h)[t];
  const float4 b  = ((const float4*)bias)[lane];
  const float4 wh = ((const float4*)Wh)[lane];
  float p = 0.0f;
  p += fmaxf(a.x + w.x * di2 + b.x + hv.x, 0.0f) * wh.x;
  p += fmaxf(a.y + w.y * di2 + b.y + hv.y, 0.0f) * wh.y;
  p += fmaxf(a.z + w.z * di2 + b.z + hv.z, 0.0f) * wh.z;
  p += fmaxf(a.w + w.w * di2 + b.w + hv.w, 0.0f) * wh.w;
#pragma unroll
  for (int off = 16; off > 0; off >>= 1)
    p += __shfl_down(p, off, 32);
  if (lane == 0) out[node] = p + bh[0];
}

// ---------------------------------------------------------------------------
extern "C" void kernel_launch(void* const* d_in, const int* in_sizes, int n_in,
                              void* d_out, int out_size, void* d_ws, size_t ws_size,
                              hipStream_t stream) {
  const float* x   = (const float*)d_in[0];
  const float* W1  = (const float*)d_in[1];
  const float* b1  = (const float*)d_in[2];
  const float* W2  = (const float*)d_in[3];
  const float* b2  = (const float*)d_in[4];
  const float* Wh  = (const float*)d_in[5];
  const float* bh  = (const float*)d_in[6];
  const int*  edge = (const int*)d_in[7];
  const int*  src  = edge;            // edge_index[0]
  const int*  dst  = edge + N_EDGES;  // edge_index[1]
  float* out = (float*)d_out;

  const size_t nd = (size_t)N_NODES * DIM;
  float* A    = (float*)d_ws;  // xw / xw2
  float* B    = A + nd;        // h (layer-1 output)
  float* C    = B + nd;        // agg accumulator
  float* dinv = C + nd;        // deg -> dinv (N floats)

  hipMemsetAsync(C, 0, nd * sizeof(float), stream);
  hipMemsetAsync(dinv, 0, (size_t)N_NODES * sizeof(float), stream);

  degree_kernel<<<(N_EDGES + 255) / 256, 256, 0, stream>>>(dst, dinv, N_EDGES);
  dinv_kernel<<<(N_NODES + 255) / 256, 256, 0, stream>>>(dinv, N_NODES);

  // ----- layer 1 -----
  gemm_wmma_f32<<<N_NODES / 16, 256, 0, stream>>>(x, W1, A);
  scatter_kernel<<<N_EDGES / 8, 256, 0, stream>>>(A, src, dst, dinv, C);
  combine1_kernel<<<(N_NODES * 32) / 256, 256, 0, stream>>>(C, A, x, b1, dinv, B);

  // ----- layer 2 + head -----
  gemm_wmma_f32<<<N_NODES / 16, 256, 0, stream>>>(B, W2, A);
  scatter_kernel<<<N_EDGES / 8, 256, 0, stream>>>(A, src, dst, dinv, C);
  combine2_head_kernel<<<N_NODES / 8, 256, 0, stream>>>(C, A, B, b2, dinv, Wh, bh, out);
}